// YoloLoss_22368189678034
// MI455X (gfx1250) — compile-verified
//
#include <hip/hip_runtime.h>

// ---------------- problem constants (from reference) ----------------
constexpr int   kC    = 25;     // 5 + NUM_CLASSES channels
constexpr int   kG2   = 49;     // 7x7 grid cells
constexpr int   kGW   = 7;      // grid width
constexpr int   kN    = 16;     // boxes per image
constexpr int   kNCls = 20;

constexpr int IMG_PER_BLOCK = 8;
constexpr int THREADS       = 256;                    // 8 wave32s
constexpr int SLAB          = kC * kG2;               // 1225 floats per image
constexpr int BLK_FLOATS    = IMG_PER_BLOCK * SLAB;   // 9800 floats = 39200 B (16B multiple)
constexpr int B128_CHUNKS   = BLK_FLOATS / 4;         // 2450 x 16B async chunks

// CDNA5 async global->LDS path (ASYNCcnt), guarded so the file always compiles.
#if defined(__AMDGCN__) && defined(__gfx1250__) && __has_builtin(__builtin_amdgcn_global_load_async_to_lds_b128)
#define USE_ASYNC 1
#else
#define USE_ASYNC 0
#endif

// builtin parameter type (from hipcc diagnostic): int vector_size(16), global/LDS AS
typedef int v4i_t __attribute__((vector_size(4 * sizeof(int))));
typedef __attribute__((address_space(1))) v4i_t* gptr_v4i;
typedef __attribute__((address_space(3))) v4i_t* lptr_v4i;

// =====================================================================
// Kernel 1: per-block partial sums of (bce_sum, box_sum, class_sum)
// =====================================================================
extern "C" __global__ void __launch_bounds__(THREADS)
yolo_loss_main(const float* __restrict__ outputs,   // [B, 25, 7, 7]
               const float* __restrict__ boxes,     // [B, 16, 4]
               const int*   __restrict__ labels,    // [B, 16] (1-indexed)
               const int*   __restrict__ coords,    // [B, 16, 2] (x, y)
               const float* __restrict__ objm,      // [B, 7, 7]
               float* __restrict__ partial,         // [nblk, 3]
               int nimg)
{
    __shared__ float lds[BLK_FLOATS];
    __shared__ float wr[3][THREADS / 32];

    const int tid   = threadIdx.x;
    const int img0  = blockIdx.x * IMG_PER_BLOCK;
    int nblkimg     = nimg - img0;
    if (nblkimg > IMG_PER_BLOCK) nblkimg = IMG_PER_BLOCK;
    const float* src = outputs + (size_t)img0 * SLAB;

    // ---- Phase A: kick off async DMA of the 8 image slabs (39.2 KB) ----
#if USE_ASYNC
    const bool full = (nblkimg == IMG_PER_BLOCK);
    if (full) {
        for (int chunk = tid; chunk < B128_CHUNKS; chunk += THREADS) {
            __builtin_amdgcn_global_load_async_to_lds_b128(
                (gptr_v4i)(src + 4 * chunk),
                (lptr_v4i)(&lds[4 * chunk]),
                /*offset=*/0, /*cpol=*/0);
        }
    } else {
        for (int i = tid; i < nblkimg * SLAB; i += THREADS) lds[i] = src[i];
    }
#else
    for (int i = tid; i < nblkimg * SLAB; i += THREADS) lds[i] = src[i];
#endif

    // ---- Phase B0: prefetch LDS-independent operands, overlapping the DMA ----
    // obj_matrix is contiguous across the block's images: objm[img0*49 + i]
    const int nterms = nblkimg * kG2;                 // <= 392, so at most 2 per thread
    const int iA = tid, iB = tid + THREADS;
    float tA = 0.0f, tB = 0.0f;
    if (iA < nterms) tA = objm[(size_t)img0 * kG2 + iA];
    if (iB < nterms) tB = objm[(size_t)img0 * kG2 + iB];

    int2   xy  = make_int2(0, 0);
    float4 tb  = make_float4(0.f, 0.f, 0.f, 0.f);
    int    lab = 0;
    const bool hasBox = (tid < nblkimg * kN);
    if (hasBox) {
        const int img  = tid >> 4;          // kN == 16
        const int n    = tid & 15;
        const int gimg = img0 + img;
        xy  = ((const int2*)coords)[(size_t)gimg * kN + n];   // (x, y)
        tb  = ((const float4*)boxes)[(size_t)gimg * kN + n];
        lab = labels[(size_t)gimg * kN + n] - 1;
    }

    // ---- drain DMA, then make slabs visible block-wide ----
#if USE_ASYNC
    if (full) {
#if __has_builtin(__builtin_amdgcn_s_wait_asynccnt)
        __builtin_amdgcn_s_wait_asynccnt(0);
#else
        asm volatile("s_wait_asynccnt 0" ::: "memory");
#endif
    }
#endif
    __syncthreads();

    // ---- Phase B1: objectness BCE terms (channel 0 vs obj_matrix) ----
    float bce = 0.0f;
    if (iA < nterms) {
        const int img  = iA / kG2;
        const int cell = iA - img * kG2;
        const float p  = lds[img * SLAB + cell];
        bce += tA * fmaxf(logf(p), -100.0f)
             + (1.0f - tA) * fmaxf(logf(1.0f - p), -100.0f);
    }
    if (iB < nterms) {
        const int img  = iB / kG2;
        const int cell = iB - img * kG2;
        const float p  = lds[img * SLAB + cell];
        bce += tB * fmaxf(logf(p), -100.0f)
             + (1.0f - tB) * fmaxf(logf(1.0f - p), -100.0f);
    }

    // ---- Phase B2: per-box MSE + class CE (one thread per box) ----
    float boxs = 0.0f, clss = 0.0f;
    if (hasBox) {
        const int img = tid >> 4;
        const int idx = xy.y * kGW + xy.x;
        const float* cellp = &lds[img * SLAB + idx];

        // box MSE: target = round(box / 224 * 10) / 10  (round half-to-even)
        const float t0 = rintf(tb.x / 224.0f * 10.0f) / 10.0f;
        const float t1 = rintf(tb.y / 224.0f * 10.0f) / 10.0f;
        const float t2 = rintf(tb.z / 224.0f * 10.0f) / 10.0f;
        const float t3 = rintf(tb.w / 224.0f * 10.0f) / 10.0f;
        const float d0 = cellp[1 * kG2] - t0;
        const float d1 = cellp[2 * kG2] - t1;
        const float d2 = cellp[3 * kG2] - t2;
        const float d3 = cellp[4 * kG2] - t3;
        boxs = (d0 * d0 + d1 * d1 + d2 * d2 + d3 * d3) * 0.25f;

        // class CE: lse - logit[label-1], max-subtracted like jax log_softmax
        float l[kNCls];
        float mx = -3.402823466e+38f;
        #pragma unroll
        for (int j = 0; j < kNCls; ++j) {
            l[j] = cellp[(5 + j) * kG2];
            mx   = fmaxf(mx, l[j]);
        }
        float s = 0.0f;
        #pragma unroll
        for (int j = 0; j < kNCls; ++j) s += expf(l[j] - mx);
        clss = (mx + logf(s)) - l[lab];
    }

    // ---- deterministic block reduction: wave32 shuffle + LDS ----
    float v0 = bce, v1 = boxs, v2 = clss;
    for (int off = 16; off > 0; off >>= 1) {
        v0 += __shfl_down(v0, off, 32);
        v1 += __shfl_down(v1, off, 32);
        v2 += __shfl_down(v2, off, 32);
    }
    const int wave = tid >> 5, lane = tid & 31;
    if (lane == 0) { wr[0][wave] = v0; wr[1][wave] = v1; wr[2][wave] = v2; }
    __syncthreads();
    if (tid == 0) {
        float s0 = 0.0f, s1 = 0.0f, s2 = 0.0f;
        for (int w = 0; w < THREADS / 32; ++w) { s0 += wr[0][w]; s1 += wr[1][w]; s2 += wr[2][w]; }
        partial[blockIdx.x * 3 + 0] = s0;
        partial[blockIdx.x * 3 + 1] = s1;
        partial[blockIdx.x * 3 + 2] = s2;
    }
}

// =====================================================================
// Kernel 2: fixed-order final reduction (deterministic), f64 accumulate
// =====================================================================
extern "C" __global__ void __launch_bounds__(256)
yolo_loss_reduce(const float* __restrict__ partial, int nblk, int nimg,
                 float* __restrict__ out)
{
    __shared__ double r[3][256];
    const int tid = threadIdx.x;
    double s0 = 0.0, s1 = 0.0, s2 = 0.0;
    for (int i = tid; i < nblk; i += 256) {
        s0 += (double)partial[i * 3 + 0];
        s1 += (double)partial[i * 3 + 1];
        s2 += (double)partial[i * 3 + 2];
    }
    r[0][tid] = s0; r[1][tid] = s1; r[2][tid] = s2;
    __syncthreads();
    for (int off = 128; off > 0; off >>= 1) {
        if (tid < off) {
            r[0][tid] += r[0][tid + off];
            r[1][tid] += r[1][tid + off];
            r[2][tid] += r[2][tid + off];
        }
        __syncthreads();
    }
    if (tid == 0) {
        const double cel = -r[0][0] / ((double)nimg * (double)kG2);
        const double box =  r[1][0] / (double)nimg;
        const double cls =  r[2][0] / (double)nimg;
        out[0] = (float)(cel + box + cls);
        out[1] = (float)cel;
        out[2] = (float)box;
        out[3] = (float)cls;
    }
}

// =====================================================================
extern "C" void kernel_launch(void* const* d_in, const int* in_sizes, int n_in,
                              void* d_out, int out_size, void* d_ws, size_t ws_size,
                              hipStream_t stream) {
    const float* outputs = (const float*)d_in[0];
    const float* boxes   = (const float*)d_in[1];
    const int*   labels  = (const int*)d_in[2];
    const int*   coords  = (const int*)d_in[3];
    const float* objm    = (const float*)d_in[4];

    const int nimg = in_sizes[0] / SLAB;                         // B (=16384)
    const int nblk = (nimg + IMG_PER_BLOCK - 1) / IMG_PER_BLOCK; // 2048

    float* partial = (float*)d_ws;                               // nblk*3 floats

    yolo_loss_main<<<nblk, THREADS, 0, stream>>>(outputs, boxes, labels, coords,
                                                 objm, partial, nimg);
    yolo_loss_reduce<<<1, 256, 0, stream>>>(partial, nblk, nimg, (float*)d_out);
}